// OverlapPatchEmbed_42013370089730
// MI455X (gfx1250) — compile-verified
//
#include <hip/hip_runtime.h>
#include <hip/hip_bf16.h>
#include <stdint.h>

// OverlapPatchEmbed: x (64,3,224,224) f32, patch=16, stride=8 -> out (64,729,3,256) f32.
// Pure data movement: ~38.5MB read + 143MB write => HBM-write bound (~7.8us @ 23.3TB/s).
// Strategy: per-(b,c,py) workgroup DMAs the 16 contiguous input rows (14,336B) into LDS
// via the CDNA5 Tensor Data Mover, then streams 27 patches out with coalesced
// non-temporal 128-bit stores (output is never re-read; keep L2 for the input).

#define BATCH 64
#define CHANS 3
#define HW    224
#define PATCH 16
#define STRIDE 8
#define NPATCH 27   // (224-16)/8 + 1

typedef uint32_t u32x4 __attribute__((ext_vector_type(4)));
typedef int      i32x4 __attribute__((ext_vector_type(4)));
typedef int      i32x8 __attribute__((ext_vector_type(8)));
typedef float    f32x4 __attribute__((ext_vector_type(4)));   // native vector: valid for nontemporal builtins

__global__ __launch_bounds__(256) void overlap_patch_kernel(const float* __restrict__ x,
                                                            float* __restrict__ out) {
    // 16 rows x 224 cols of f32: one contiguous 14,336B band of the input plane.
    __shared__ __align__(16) float tile[PATCH * HW];

    const int blk = blockIdx.x;
    const int b   = blk / (CHANS * NPATCH);
    const int rem = blk - b * (CHANS * NPATCH);
    const int c   = rem / NPATCH;
    const int py  = rem - c * NPATCH;

    const float* src = x + ((size_t)(b * CHANS + c) * (HW * HW) + (size_t)py * STRIDE * HW);

#if __has_builtin(__builtin_amdgcn_tensor_load_to_lds)
    if (threadIdx.x < 32) {  // wave 0 issues the TDM descriptor once per workgroup
        const uint64_t gaddr = (uint64_t)(uintptr_t)src;
        const uint32_t laddr = (uint32_t)(uintptr_t)tile;  // low 32 bits = LDS byte offset
        const uint32_t dim0  = PATCH * HW;                 // 3584 f32 elements, 1-D tile

        u32x4 g0;
        g0[0] = 1u;                                              // count=1 (valid user D#)
        g0[1] = laddr;                                           // lds_addr (bytes)
        g0[2] = (uint32_t)(gaddr & 0xFFFFFFFFull);               // global_addr[31:0]
        g0[3] = (uint32_t)((gaddr >> 32) & 0x01FFFFFFull)        // global_addr[56:32]
              | (2u << 30);                                      // type = 2 ("image")

        i32x8 g1;
        g1[0] = (int)(2u << 16);                                 // data_size = 4 bytes
        g1[1] = (int)((dim0 & 0xFFFFu) << 16);                   // tensor_dim0[15:0]
        g1[2] = (int)(((dim0 >> 16) & 0xFFFFu) | (1u << 16));    // tensor_dim0[31:16] | tensor_dim1=1
        g1[3] = (int)((dim0 & 0xFFFFu) << 16);                   // tensor_dim1 hi=0 | tile_dim0=3584
        g1[4] = 1;                                               // tile_dim1=1, tile_dim2=0 (unused)
        g1[5] = (int)dim0;                                       // tensor_dim0_stride[31:0]
        g1[6] = 0;                                               // stride0 hi | stride1 lo
        g1[7] = 0;                                               // stride1 hi

        // Groups 2/3: higher dims forced to extent 1 so no OOB path can trigger.
        i32x4 g2; g2[0] = 1; g2[1] = 1; g2[2] = 0; g2[3] = 0;    // tensor_dim2=1, tensor_dim3=1
        i32x4 g3; g3[0] = 0; g3[1] = (int)(1u << 16); g3[2] = 0; g3[3] = 0; // tensor_dim4=1

#if defined(__clang_major__) && (__clang_major__ >= 23)
        i32x8 gz8 = {0, 0, 0, 0, 0, 0, 0, 0};
        __builtin_amdgcn_tensor_load_to_lds(g0, g1, g2, g3, gz8, 0);
#else
        __builtin_amdgcn_tensor_load_to_lds(g0, g1, g2, g3, 0);
#endif
        __builtin_amdgcn_s_wait_tensorcnt(0);
    }
#else
    // Fallback: cooperative 128-bit loads into LDS (source band is contiguous & 16B-aligned).
    {
        const f32x4* s4 = (const f32x4*)src;
        f32x4*       t4 = (f32x4*)tile;
        for (int k = threadIdx.x; k < (PATCH * HW) / 4; k += 256) t4[k] = s4[k];
    }
#endif
    __syncthreads();

    // Emit 27 patches x 256 floats. Per (patch,c) the 256 floats are contiguous;
    // consecutive lanes write consecutive float4s -> coalesced 1KB bursts.
    f32x4* outv = (f32x4*)out;
    const size_t obase = ((size_t)b * (NPATCH * NPATCH) + (size_t)py * NPATCH) * CHANS + c;

    const int tid = threadIdx.x;
    #pragma unroll
    for (int k = 0; k < 7; ++k) {                 // 27*64 = 1728 float4s / 256 threads
        const int idx = k * 256 + tid;
        if (idx < NPATCH * 64) {
            const int px = idx >> 6;              // patch column 0..26
            const int w  = idx & 63;              // float4 within the 256-float patch
            const int i  = w >> 2;                // patch row 0..15
            const int jv = w & 3;                 // float4 within the 16-float row
            // LDS: row i of band, columns [px*8 + jv*4 .. +3] -- 16B aligned.
            const f32x4 v = *(const f32x4*)&tile[i * HW + px * STRIDE + jv * 4];
            const size_t o = (obase + (size_t)px * CHANS) * 64 + (size_t)(i * 4 + jv);
            __builtin_nontemporal_store(v, &outv[o]);
        }
    }
}

extern "C" void kernel_launch(void* const* d_in, const int* in_sizes, int n_in,
                              void* d_out, int out_size, void* d_ws, size_t ws_size,
                              hipStream_t stream) {
    const float* x = (const float*)d_in[0];
    float* out = (float*)d_out;
    (void)in_sizes; (void)n_in; (void)out_size; (void)d_ws; (void)ws_size;

    dim3 grid(BATCH * CHANS * NPATCH);   // 5184 workgroups
    dim3 block(256);                     // 8 wave32s; ~14KB LDS -> high occupancy
    overlap_patch_kernel<<<grid, block, 0, stream>>>(x, out);
}